// ConvSV_11682311045197
// MI455X (gfx1250) — compile-verified
//
#include <hip/hip_runtime.h>
#include <hip/hip_bf16.h>

typedef __attribute__((ext_vector_type(2))) float v2f;
typedef __attribute__((ext_vector_type(8))) float v8f;

// Per-atom: avf_sv[16x64] = A_jT(16x64-K) @ g_sv(64x64), via V_WMMA_F32_16X16X4_F32.
// Block = 128 threads (4 waves, wave32). Wave w computes the 16-col tile
// [16w, 16w+16) of the gd=64 dimension. Then stage-2 contraction on VALU.
__global__ void __launch_bounds__(128)
ConvSV_kernel(const float* __restrict__ A,      // (N,16)
              const int*   __restrict__ NB2,    // (N,64) int64 viewed as int32 pairs
              const float* __restrict__ GSV,    // (N,64,64)
              const float* __restrict__ AGH,    // (16,16,16)
              float*       __restrict__ OUT)    // (N,512)
{
    __shared__ int   s_nb[64];
    __shared__ float s_sv[16 * 68];   // 16 rows (channel a) x 64 cols (g*4+d), stride 68 pad

    const int n    = blockIdx.x;
    const int tid  = threadIdx.x;
    const int lane = tid & 31;
    const int wave = tid >> 5;

    // Load neighbor indices (low dword of each int64) into LDS.
    if (tid < 64) s_nb[tid] = NB2[((size_t)n * 64 + tid) * 2];
    __syncthreads();

    const int half = lane >> 4;     // 0: lanes 0-15, 1: lanes 16-31
    const int lan  = lane & 15;

    // g_sv row base for this wave's 16-column tile; rows are 64 floats apart.
    const float* gbase = GSV + (size_t)n * 64 * 64 + wave * 16 + lan;

    v8f c = {0.f, 0.f, 0.f, 0.f, 0.f, 0.f, 0.f, 0.f};

    // K (= neighbor index m) in steps of 4.
    #pragma unroll 4
    for (int k0 = 0; k0 < 64; k0 += 4) {
        const int ka = k0 + half * 2;       // half-wave owns K = ka, ka+1
        const int j0 = s_nb[ka];
        const int j1 = s_nb[ka + 1];

        // A fragment (16x4, M=channel, K=neighbor): A[M=lan][K] = a[nb[K]][lan]
        v2f af, bf;
        af.x = A[(size_t)j0 * 16 + lan];
        af.y = A[(size_t)j1 * 16 + lan];
        // B fragment (4x16, K=neighbor, N=gd col): B[K][N=lan] = g_sv[n][K][16w+lan]
        bf.x = gbase[(size_t)ka * 64];
        bf.y = gbase[(size_t)(ka + 1) * 64];

        c = __builtin_amdgcn_wmma_f32_16x16x4_f32(
                /*neg_a=*/false, af, /*neg_b=*/false, bf,
                /*c_mod=*/(short)0, c, /*reuse_a=*/false, /*reuse_b=*/false);
    }

    // Scatter the 16x16 f32 tile to LDS.
    // VGPR i holds M = i (lanes 0-15) / M = i+8 (lanes 16-31); N = lan.
    const int colg  = wave * 16 + lan;
    const int mbase = half * 8;
    #pragma unroll
    for (int i = 0; i < 8; ++i)
        s_sv[(mbase + i) * 68 + colg] = c[i];
    __syncthreads();

    const size_t obase = (size_t)n * 512;

    // avf_s output: out[n, a*16 + g] = sv[a][g*4 + 0]
    #pragma unroll
    for (int idx = tid; idx < 256; idx += 128) {
        const int aa = idx >> 4, g = idx & 15;
        OUT[obase + idx] = s_sv[aa * 68 + g * 4];
    }

    // avf_v: per (a,h): sum_d=1..3 ( sum_g agh[a,g,h] * sv[a][g*4+d] )^2
    const int aa = tid >> 3;        // channel 0..15
    const int h0 = tid & 7;         // this thread does h0 and h0+8
    #pragma unroll
    for (int hh = 0; hh < 2; ++hh) {
        const int h = h0 + hh * 8;
        float acc = 0.f;
        #pragma unroll
        for (int d = 1; d < 4; ++d) {
            float t = 0.f;
            #pragma unroll
            for (int g = 0; g < 16; ++g)
                t += AGH[(aa * 16 + g) * 16 + h] * s_sv[aa * 68 + g * 4 + d];
            acc += t * t;
        }
        OUT[obase + 256 + aa * 16 + h] = acc;
    }
}

extern "C" void kernel_launch(void* const* d_in, const int* in_sizes, int n_in,
                              void* d_out, int out_size, void* d_ws, size_t ws_size,
                              hipStream_t stream) {
    const float* A   = (const float*)d_in[0];   // a: (N,16) f32
    const int*   NB2 = (const int*)d_in[1];     // nbmat: (N,64) int64 -> int32 pairs
    const float* GSV = (const float*)d_in[2];   // g_sv: (N,64,16,4) f32
    const float* AGH = (const float*)d_in[3];   // agh: (16,16,16) f32
    float*       OUT = (float*)d_out;           // (N,512) f32

    const int N = in_sizes[0] / 16;             // N_ATOMS
    ConvSV_kernel<<<N, 128, 0, stream>>>(A, NB2, GSV, AGH, OUT);
}